// DH_SFNN_86474871538197
// MI455X (gfx1250) — compile-verified
//
#include <hip/hip_runtime.h>

typedef float v2f __attribute__((ext_vector_type(2)));
typedef float v4f __attribute__((ext_vector_type(4)));
typedef float v8f __attribute__((ext_vector_type(8)));

// Problem constants (from reference)
constexpr int Bsz  = 256;
constexpr int T    = 500;
constexpr int DIN  = 700;
constexpr int H    = 200;
constexpr int BR   = 2;
constexpr int NOUT = H * BR;   // 400
constexpr int DOUT = 35;

// Chunking of the time axis so the branch-current intermediate fits in ws.
constexpr int TC   = 100;          // timesteps per chunk (500 / 100 = 5 chunks)
constexpr int MCH  = Bsz * TC;     // 25600 GEMM rows per chunk
constexpr int NT   = NOUT / 16;    // 25 N-tiles
constexpr int MT4  = MCH / 64;     // 400 M-supertiles (64 rows each)
constexpr int GEMM_WAVES  = NT * MT4;        // 10000 waves
constexpr int GEMM_BLOCKS = GEMM_WAVES / 8;  // 1250 blocks of 256 threads (8 waves)

constexpr int KG   = 8;            // K per group (2 WMMA steps per b128 load set)
constexpr int NG   = DIN / KG;     // 87 full groups (k = 0..695)
constexpr int KTAIL = DIN - NG * KG; // 4 remaining K (696..699)

__device__ __forceinline__ float sigmoidf_(float x) {
    return 1.0f / (1.0f + __expf(-x));
}

#define WMMA_F32(A, B, C) \
    __builtin_amdgcn_wmma_f32_16x16x4_f32(false, (A), false, (B), (short)0, (C), false, false)

__device__ __forceinline__ v2f lo2_(v4f v) { return __builtin_shufflevector(v, v, 0, 1); }
__device__ __forceinline__ v2f hi2_(v4f v) { return __builtin_shufflevector(v, v, 2, 3); }

// ---------------------------------------------------------------------------
// Kernel 1: branch_in chunk GEMM via V_WMMA_F32_16X16X4_F32 (fp32-exact).
// Per 8-K group: 5 x global_load_b128 (1 B + 4 A fragments) feed 8 WMMAs.
// K-permutation inside a group: WMMA s0 covers k-set {k0,k0+1,k0+4,k0+5}
// (halo0 lanes -> k0,k0+1; halo1 lanes -> k0+4,k0+5), s1 covers the rest;
// A and B use the same permutation so the product is exact.
// Software-pipelined: next group's loads are issued before the current
// group's 8 WMMAs.
// ---------------------------------------------------------------------------
__global__ __launch_bounds__(256) void dhsfnn_gemm_chunk(
    const float* __restrict__ x, const float* __restrict__ Win,
    const float* __restrict__ bin, float* __restrict__ outc, int t0)
{
    const int wave  = blockIdx.x * 8 + (threadIdx.x >> 5);
    const int lane  = threadIdx.x & 31;
    const int nTile = wave % NT;
    const int m4    = wave / NT;

    const int r    = lane & 15;   // row-in-tile (A) / column (B,C)
    const int halo = lane >> 4;   // upper-half-wave flag
    const int kg4  = halo * 4;    // b128 group offset: halo1 lanes take k0+4..k0+7

    // Fragment row base pointers (offset by the halo group shift)
    const float* __restrict__ bbase = Win + (size_t)(nTile * 16 + r) * DIN + kg4;
    const float* __restrict__ ab0;
    const float* __restrict__ ab1;
    const float* __restrict__ ab2;
    const float* __restrict__ ab3;
    {
        const float* ap[4];
#pragma unroll
        for (int s = 0; s < 4; ++s) {
            const int ml   = m4 * 64 + s * 16 + r;   // m_local
            const int bidx = ml / TC;
            const int tl   = ml % TC;
            ap[s] = x + ((size_t)bidx * T + (size_t)(t0 + tl)) * DIN + kg4;
        }
        ab0 = ap[0]; ab1 = ap[1]; ab2 = ap[2]; ab3 = ap[3];
    }

    v8f c0 = {}, c1 = {}, c2 = {}, c3 = {};

    // ---- software pipeline: group 0 prefetch (5 x b128) ----
    v4f bB = *(const v4f*)(bbase);
    v4f aA0 = *(const v4f*)(ab0);
    v4f aA1 = *(const v4f*)(ab1);
    v4f aA2 = *(const v4f*)(ab2);
    v4f aA3 = *(const v4f*)(ab3);

    // steady state: 86 iterations (even -> unroll 2 renames the rotation)
#pragma unroll 2
    for (int k0 = KG; k0 < NG * KG; k0 += KG) {
        const v4f bBn = *(const v4f*)(bbase + k0);
        const v4f a0n = *(const v4f*)(ab0 + k0);
        const v4f a1n = *(const v4f*)(ab1 + k0);
        const v4f a2n = *(const v4f*)(ab2 + k0);
        const v4f a3n = *(const v4f*)(ab3 + k0);

        const v2f bl = lo2_(bB), bh = hi2_(bB);
        c0 = WMMA_F32(lo2_(aA0), bl, c0);
        c1 = WMMA_F32(lo2_(aA1), bl, c1);
        c2 = WMMA_F32(lo2_(aA2), bl, c2);
        c3 = WMMA_F32(lo2_(aA3), bl, c3);
        c0 = WMMA_F32(hi2_(aA0), bh, c0);
        c1 = WMMA_F32(hi2_(aA1), bh, c1);
        c2 = WMMA_F32(hi2_(aA2), bh, c2);
        c3 = WMMA_F32(hi2_(aA3), bh, c3);

        bB = bBn; aA0 = a0n; aA1 = a1n; aA2 = a2n; aA3 = a3n;
    }

    // consume the final full group
    {
        const v2f bl = lo2_(bB), bh = hi2_(bB);
        c0 = WMMA_F32(lo2_(aA0), bl, c0);
        c1 = WMMA_F32(lo2_(aA1), bl, c1);
        c2 = WMMA_F32(lo2_(aA2), bl, c2);
        c3 = WMMA_F32(lo2_(aA3), bl, c3);
        c0 = WMMA_F32(hi2_(aA0), bh, c0);
        c1 = WMMA_F32(hi2_(aA1), bh, c1);
        c2 = WMMA_F32(hi2_(aA2), bh, c2);
        c3 = WMMA_F32(hi2_(aA3), bh, c3);
    }

    // ---- K tail: 4 remaining k (696..699), classic b64 fragments ----
    {
        const int ktail = NG * KG;                 // 696
        const int shift = halo * 2 - kg4;          // rebase from kg4 to kofs=halo*2
        const v2f bt = *(const v2f*)(bbase + ktail + shift);
        const v2f t0v = *(const v2f*)(ab0 + ktail + shift);
        const v2f t1v = *(const v2f*)(ab1 + ktail + shift);
        const v2f t2v = *(const v2f*)(ab2 + ktail + shift);
        const v2f t3v = *(const v2f*)(ab3 + ktail + shift);
        c0 = WMMA_F32(t0v, bt, c0);
        c1 = WMMA_F32(t1v, bt, c1);
        c2 = WMMA_F32(t2v, bt, c2);
        c3 = WMMA_F32(t3v, bt, c3);
    }

    const float bias = bin[nTile * 16 + r];
    const int   col  = nTile * 16 + r;

#pragma unroll
    for (int j = 0; j < 8; ++j) {
        const int rowj = j + 8 * halo;  // C layout: VGPR j -> M=j (lo half) / j+8 (hi half)
        outc[(size_t)(m4 * 64 + 0 * 16 + rowj) * NOUT + col] = c0[j] + bias;
        outc[(size_t)(m4 * 64 + 1 * 16 + rowj) * NOUT + col] = c1[j] + bias;
        outc[(size_t)(m4 * 64 + 2 * 16 + rowj) * NOUT + col] = c2[j] + bias;
        outc[(size_t)(m4 * 64 + 3 * 16 + rowj) * NOUT + col] = c3[j] + bias;
    }
}

// ---------------------------------------------------------------------------
// Kernel 2: LIF scan over one TC-timestep chunk. One thread per (b,h).
// Persistent state (v, i_d[2], counts) lives in workspace across chunks.
// ---------------------------------------------------------------------------
__global__ __launch_bounds__(256) void dhsfnn_scan_chunk(
    const float* __restrict__ branch, const float* __restrict__ tau_n,
    const float* __restrict__ tau_m, float* __restrict__ v_state,
    float* __restrict__ id_state, float* __restrict__ counts)
{
    const int idx = blockIdx.x * blockDim.x + threadIdx.x;  // 0 .. B*H-1
    const int b = idx / H;
    const int h = idx % H;

    const float beta0 = sigmoidf_(tau_n[h * BR + 0]);
    const float beta1 = sigmoidf_(tau_n[h * BR + 1]);
    const float alpha = sigmoidf_(tau_m[h]);
    const float ob0 = 1.0f - beta0, ob1 = 1.0f - beta1, oa = 1.0f - alpha;

    float v   = v_state[idx];
    float i0  = id_state[idx * 2 + 0];
    float i1  = id_state[idx * 2 + 1];
    float cnt = counts[idx];

    const float* __restrict__ base = branch + (size_t)b * TC * NOUT + h * BR;
    for (int tl = 0; tl < TC; ++tl) {
        const v2f inp = *(const v2f*)(base + (size_t)tl * NOUT);
        i0 = beta0 * i0 + ob0 * inp[0];
        i1 = beta1 * i1 + ob1 * inp[1];
        v  = alpha * v + oa * (i0 + i1);
        if (v >= 1.0f) { v -= 1.0f; cnt += 1.0f; }
    }

    v_state[idx]          = v;
    id_state[idx * 2 + 0] = i0;
    id_state[idx * 2 + 1] = i1;
    counts[idx]           = cnt;
}

// ---------------------------------------------------------------------------
// Kernel 3: output projection  out[b,o] = sum_h counts[b,h]*W_out[o,h] + b_out[o]
// ---------------------------------------------------------------------------
__global__ __launch_bounds__(256) void dhsfnn_out_proj(
    const float* __restrict__ counts, const float* __restrict__ Wout,
    const float* __restrict__ bout, float* __restrict__ out)
{
    const int idx = blockIdx.x * blockDim.x + threadIdx.x;
    if (idx >= Bsz * DOUT) return;
    const int b = idx / DOUT;
    const int o = idx % DOUT;
    float acc = bout[o];
    const float* __restrict__ cr = counts + (size_t)b * H;
    const float* __restrict__ wr = Wout + (size_t)o * H;
    for (int h = 0; h < H; ++h) acc += cr[h] * wr[h];
    out[(size_t)b * DOUT + o] = acc;
}

// Zero persistent scan state at the start of every launch (determinism).
__global__ void dhsfnn_zero(float* __restrict__ p, int n)
{
    const int i = blockIdx.x * blockDim.x + threadIdx.x;
    if (i < n) p[i] = 0.0f;
}

extern "C" void kernel_launch(void* const* d_in, const int* in_sizes, int n_in,
                              void* d_out, int out_size, void* d_ws, size_t ws_size,
                              hipStream_t stream) {
    const float* x     = (const float*)d_in[0];
    const float* W_in  = (const float*)d_in[1];
    const float* b_in  = (const float*)d_in[2];
    const float* tau_n = (const float*)d_in[3];
    const float* tau_m = (const float*)d_in[4];
    const float* W_out = (const float*)d_in[5];
    const float* b_out = (const float*)d_in[6];
    float* out = (float*)d_out;

    // Workspace layout
    const size_t branchBytes = (size_t)MCH * NOUT * sizeof(float); // 40,960,000 B
    float* ws_branch = (float*)d_ws;
    float* v_state   = (float*)((char*)d_ws + branchBytes);
    float* id_state  = v_state + (size_t)Bsz * H;
    float* counts    = id_state + (size_t)Bsz * H * 2;

    // Zero (v, i_d, counts): 4 * B * H floats contiguous starting at v_state
    const int stateN = Bsz * H * 4;
    dhsfnn_zero<<<(stateN + 255) / 256, 256, 0, stream>>>(v_state, stateN);

    for (int c = 0; c < T / TC; ++c) {
        dhsfnn_gemm_chunk<<<GEMM_BLOCKS, 256, 0, stream>>>(
            x, W_in, b_in, ws_branch, c * TC);
        dhsfnn_scan_chunk<<<(Bsz * H) / 256, 256, 0, stream>>>(
            ws_branch, tau_n, tau_m, v_state, id_state, counts);
    }

    dhsfnn_out_proj<<<(Bsz * DOUT + 255) / 256, 256, 0, stream>>>(
        counts, W_out, b_out, out);
}